// GraphConv_21689584844830
// MI455X (gfx1250) — compile-verified
//
#include <hip/hip_runtime.h>

#define N_NODES 50000
#define N_EDGES 600000
#define FDIM    128          // F_IN == F_OUT == 128
#define LDS_STRIDE 132       // padded row stride (floats); 132*4 B = 528 = 33*16 -> float4-aligned rows

typedef __attribute__((ext_vector_type(2))) float v2f;
typedef __attribute__((ext_vector_type(8))) float v8f;

// ---------------------------------------------------------------------------
// Kernel 1: zero the accumulator (d_out is used as the segment-sum buffer,
// then overwritten in place by the GEMM). float4 stores -> global_store_b128.
// ---------------------------------------------------------------------------
__global__ void __launch_bounds__(256)
zero_kernel(float4* __restrict__ p, int n4) {
    int i = blockIdx.x * blockDim.x + threadIdx.x;
    if (i < n4) p[i] = make_float4(0.f, 0.f, 0.f, 0.f);
}

// ---------------------------------------------------------------------------
// Kernel 2: edge scatter-add.  One wave (32 lanes) per edge; each lane moves
// one float4 chunk of the 128-float feature row.  Gather is global_load_b128
// (x is L2-resident: 25.6 MB << 192 MB L2); accumulation uses hardware
// GLOBAL_ATOMIC_ADD_F32 without return (STOREcnt path, resolved at L2).
// ---------------------------------------------------------------------------
__global__ void __launch_bounds__(256)
scatter_kernel(const float4* __restrict__ x,   // [N, 32] as float4
               const int*    __restrict__ eidx,// [2, E]
               float*        __restrict__ acc) // [N, 128]
{
    unsigned gid = blockIdx.x * blockDim.x + threadIdx.x;
    unsigned e = gid >> 5;          // edge id (grid sized exactly: no check)
    unsigned q = gid & 31;          // float4 chunk within feature row
    int src = eidx[e];
    int dst = eidx[N_EDGES + e];
    float4 v = x[(size_t)src * (FDIM / 4) + q];
    float* d = acc + (size_t)dst * FDIM + q * 4;
    unsafeAtomicAdd(d + 0, v.x);
    unsafeAtomicAdd(d + 1, v.y);
    unsafeAtomicAdd(d + 2, v.z);
    unsafeAtomicAdd(d + 3, v.w);
}

// ---------------------------------------------------------------------------
// Kernel 3: in-place GEMM  acc[16x128] @ W2[128x128] + b2  -> same 16 rows.
// Block = 256 threads (8 waves) = one 16-row M-block.  A tile staged to LDS
// (padded stride -> conflict-free ds_load_b64 fragment reads), each wave owns
// one 16-column N-tile and runs 32 x V_WMMA_F32_16X16X4_F32 (exact f32).
// Dependency is block-diagonal in M, so writing back over acc is safe after
// the staging barrier.
// ---------------------------------------------------------------------------
__global__ void __launch_bounds__(256)
gemm_kernel(float*       __restrict__ acc_out, // [N,128] in/out (d_out)
            const float* __restrict__ W2,      // [128,128] row-major
            const float* __restrict__ b2)      // [128]
{
    __shared__ __align__(16) float lds[16 * LDS_STRIDE];

    const int tid    = threadIdx.x;
    const int wave   = tid >> 5;
    const int lane   = tid & 31;
    const int m_base = blockIdx.x * 16;

    // ---- cooperative stage of the 16x128 A tile into LDS (2 float4 / thread)
    const float4* arow = (const float4*)(acc_out + (size_t)m_base * FDIM);
    #pragma unroll
    for (int i = 0; i < 2; ++i) {
        int idx = tid + i * 256;            // float4 index in [0,512)
        int r   = idx >> 5;                 // row   0..15
        int c4  = idx & 31;                 // col/4 0..31
        float4 g = arow[idx];
        *(float4*)(&lds[r * LDS_STRIDE + c4 * 4]) = g;  // ds_store_b128
    }
    __syncthreads();

    const int n_base = wave * 16;
    const int half   = lane >> 4;           // 0 | 1  (K sub-group)
    const int l16    = lane & 15;

    // C/D layout: VGPR v, lanes 0-15 -> M=v, lanes 16-31 -> M=v+8; N = l16.
    // Preload bias into the accumulator (bias is constant along M).
    v8f c;
    {
        float bias = b2[n_base + l16];
        #pragma unroll
        for (int v = 0; v < 8; ++v) c[v] = bias;
    }

    // A frag: lane holds A[m = l16][k = k0 + 2*half + {0,1}]  (ds_load_b64)
    // B frag: lane holds B[k = k0 + 2*half + {0,1}][n = n_base + l16]
    const float* a_ptr = &lds[l16 * LDS_STRIDE + half * 2];
    const float* b_ptr = &W2[(size_t)(half * 2) * FDIM + n_base + l16];

    #pragma unroll
    for (int k0 = 0; k0 < FDIM; k0 += 4) {
        v2f a, b;
        a.x = a_ptr[k0 + 0];
        a.y = a_ptr[k0 + 1];
        b.x = b_ptr[(size_t)(k0 + 0) * FDIM];
        b.y = b_ptr[(size_t)(k0 + 1) * FDIM];
        // D = A x B + C   (v_wmma_f32_16x16x4_f32)
        c = __builtin_amdgcn_wmma_f32_16x16x4_f32(
                /*neg_a=*/false, a, /*neg_b=*/false, b,
                /*c_mod=*/(short)0, c, /*reuse_a=*/false, /*reuse_b=*/false);
    }

    // ---- write back in place (LDS copy already holds A; no 2nd barrier)
    float* out_ptr = acc_out + (size_t)(m_base + half * 8) * FDIM + n_base + l16;
    #pragma unroll
    for (int v = 0; v < 8; ++v) out_ptr[(size_t)v * FDIM] = c[v];
}

// ---------------------------------------------------------------------------
// Inputs (setup_inputs order): 0:x 1:edge_index 2:edge_weight(dead) 3:W1(dead)
// 4:b1(dead) 5:W2 6:b2 7:a(dead) 8:b(dead).   Output: [50000,128] f32.
// ---------------------------------------------------------------------------
extern "C" void kernel_launch(void* const* d_in, const int* in_sizes, int n_in,
                              void* d_out, int out_size, void* d_ws, size_t ws_size,
                              hipStream_t stream) {
    (void)in_sizes; (void)n_in; (void)out_size; (void)d_ws; (void)ws_size;

    const float* x    = (const float*)d_in[0];
    const int*   eidx = (const int*)  d_in[1];
    const float* W2   = (const float*)d_in[5];
    const float* b2   = (const float*)d_in[6];
    float*       out  = (float*)d_out;

    // 1) zero the accumulator (d_out)
    const int n4 = N_NODES * FDIM / 4;                 // 1,600,000 float4
    zero_kernel<<<(n4 + 255) / 256, 256, 0, stream>>>((float4*)out, n4);

    // 2) segment-sum: one wave per edge, fp32 hardware atomics into d_out
    scatter_kernel<<<(N_EDGES * 32) / 256, 256, 0, stream>>>(
        (const float4*)x, eidx, out);

    // 3) in-place (d_out @ W2 + b2) via f32 WMMA; 3125 M-blocks x 8 waves
    gemm_kernel<<<N_NODES / 16, 256, 0, stream>>>(out, W2, b2);
}